// BinaryAggregationLayer_59304908423776
// MI455X (gfx1250) — compile-verified
//
#include <hip/hip_runtime.h>
#include <cstdint>

// BinaryAggregationLayer: out[b, j] = clip(x[b, j])             for j < 8190
//                         out[b,8190] = clip(mean(x[b,8190], x[b,8191]))
// HBM-bandwidth-bound stream (256 MiB traffic -> ~11.5 us @ 23.3 TB/s).
// No matrix work -> WMMA inapplicable. The relevant CDNA5 path is the async
// data mover: global_load_async_to_lds_b128 (HBM -> LDS, ASYNCcnt-tracked,
// th:NT so the read-once stream doesn't thrash the 192 MB L2), then an LDS
// readback, clamp, and coalesced NT dword stores.

#define IN_W   8192
#define OUT_W  8191
#define CLAMP_V 10000.0f

typedef float v4f __attribute__((ext_vector_type(4)));

__device__ __forceinline__ float clampv(float v) {
    return fminf(fmaxf(v, -CLAMP_V), CLAMP_V);
}

__global__ __launch_bounds__(256)
void ba_stream_kernel(const float* __restrict__ x, float* __restrict__ out) {
    __shared__ v4f stage[256];                 // 4 KB staging, one slot/thread

    const int row  = blockIdx.y;                                   // uniform -> SGPR
    const int col0 = (blockIdx.x * 256 + threadIdx.x) * 4;         // 4 cols / thread
    const float* __restrict__ xr = x + (size_t)row * IN_W;         // 16B-aligned row
    float* __restrict__ orow     = out + (size_t)row * OUT_W;      // 4B-aligned row

    if (col0 + 3 <= OUT_W - 2) {
        // CDNA5 async data mover: HBM -> LDS, bypassing the VGPR file.
        // Generic LDS pointer truncates to the 32-bit LDS byte offset.
        const float* gsrc  = xr + col0;
        unsigned lds_off   = (unsigned)(uintptr_t)&stage[threadIdx.x];
        unsigned long long gaddr = (unsigned long long)(uintptr_t)gsrc;

        asm volatile("global_load_async_to_lds_b128 %0, %1, off th:TH_LOAD_NT"
                     :: "v"(lds_off), "v"(gaddr)
                     : "memory");
        asm volatile("s_wait_asynccnt 0x0" ::: "memory");

        v4f v = stage[threadIdx.x];            // ds_load_b128 readback

        // Output rows are only 4B-aligned (pitch 8191*4B) -> dword NT stores,
        // still fully coalesced (wave32 dword store = one 128B line).
        __builtin_nontemporal_store(clampv(v.x), orow + col0 + 0);
        __builtin_nontemporal_store(clampv(v.y), orow + col0 + 1);
        __builtin_nontemporal_store(clampv(v.z), orow + col0 + 2);
        __builtin_nontemporal_store(clampv(v.w), orow + col0 + 3);
    } else if (col0 < OUT_W) {
        // Tail chunk (one thread per row): cols col0..8190, last one is the
        // 2-input segment mean (inputs 8190 and 8191 both route to node 8190).
        for (int j = col0; j < OUT_W; ++j) {
            float v = (j == OUT_W - 1)
                        ? 0.5f * (xr[IN_W - 2] + xr[IN_W - 1])
                        : xr[j];
            __builtin_nontemporal_store(clampv(v), orow + j);
        }
    }
}

extern "C" void kernel_launch(void* const* d_in, const int* in_sizes, int n_in,
                              void* d_out, int out_size, void* d_ws, size_t ws_size,
                              hipStream_t stream) {
    const float* x = (const float*)d_in[0];
    float* out     = (float*)d_out;

    const int batch = in_sizes[0] / IN_W;                 // 4096
    const int chunks_per_row = (OUT_W + 3) / 4;           // 2048
    dim3 block(256);
    dim3 grid((chunks_per_row + 255) / 256, batch);       // (8, 4096)

    ba_stream_kernel<<<grid, block, 0, stream>>>(x, out);
}